// GATNet_76166950027503
// MI455X (gfx1250) — compile-verified
//
#include <hip/hip_runtime.h>
#include <math.h>

#define N_NODES 65536
#define N_EDGES 1048576
#define ET (N_EDGES + N_NODES)

typedef __attribute__((ext_vector_type(2))) float v2f;
typedef __attribute__((ext_vector_type(8))) float v8f;

// ---------------------------------------------------------------------------
// Dense GEMM via V_WMMA_F32_16X16X4_F32. One wave -> one 16x16 output tile.
// A: [M,K] row-major, B: [K,Ncol] row-major, C: [M,Ncol]. K % 4 == 0.
// ---------------------------------------------------------------------------
__global__ __launch_bounds__(32)
void gemm16_wmma(const float* __restrict__ A, const float* __restrict__ B,
                 const float* __restrict__ bias, float* __restrict__ C,
                 int K, int Ncol) {
  const int tilesN = Ncol >> 4;
  const int tile = blockIdx.x;
  const int tm = tile / tilesN;
  const int tn = tile - tm * tilesN;
  const int l = threadIdx.x;
  const int half = l >> 4;        // 0 for lanes 0-15, 1 for lanes 16-31
  const int lm = l & 15;
  const float* arow = A + (size_t)(tm * 16 + lm) * K;
  const int nIdx = tn * 16 + lm;

  v8f acc = {};
  for (int k0 = 0; k0 < K; k0 += 4) {
    const int kk = k0 + 2 * half;
    v2f a, b;
    a.x = arow[kk];
    a.y = arow[kk + 1];
    b.x = B[(size_t)kk * Ncol + nIdx];
    b.y = B[(size_t)(kk + 1) * Ncol + nIdx];
    acc = __builtin_amdgcn_wmma_f32_16x16x4_f32(
        /*neg_a=*/false, a, /*neg_b=*/false, b,
        /*c_mod=*/(short)0, acc, /*reuse_a=*/false, /*reuse_b=*/false);
  }
  const float bv = bias ? bias[nIdx] : 0.0f;
#pragma unroll
  for (int j = 0; j < 8; ++j) {
    const int m = tm * 16 + j + 8 * half;
    C[(size_t)m * Ncol + nIdx] = acc[j] + bv;
  }
}

// ---------------------------------------------------------------------------
// Helpers
// ---------------------------------------------------------------------------
__global__ void fill_kernel(float* __restrict__ p, int n, float v) {
  int i = blockIdx.x * blockDim.x + threadIdx.x;
  if (i < n) p[i] = v;
}

__device__ inline void atomicMaxF(float* a, float v) {
  if (!__builtin_signbitf(v))
    atomicMax((int*)a, __float_as_int(v));
  else
    atomicMin((unsigned int*)a, __float_as_uint(v));
}

__device__ inline void edge_ends(const int* __restrict__ ei, int e, int& s, int& d) {
  if (e < N_EDGES) { s = ei[e]; d = ei[N_EDGES + e]; }
  else             { s = e - N_EDGES; d = s; }
}

// al_s[n,h] = sum_c h[n,h,c]*a_src[h,c]; al_d likewise
template <int H, int C>
__global__ void node_coeff(const float* __restrict__ h, const float* __restrict__ as,
                           const float* __restrict__ ad, float* __restrict__ als,
                           float* __restrict__ ald) {
  int i = blockIdx.x * blockDim.x + threadIdx.x;  // node*H + head
  if (i >= N_NODES * H) return;
  int n = i / H, hh = i - n * H;
  const float* hv = h + (size_t)n * H * C + hh * C;
  float s = 0.f, d = 0.f;
#pragma unroll
  for (int c = 0; c < C; ++c) {
    s += hv[c] * as[hh * C + c];
    d += hv[c] * ad[hh * C + c];
  }
  als[i] = s;
  ald[i] = d;
}

__device__ inline float leaky02(float v) { return v > 0.f ? v : 0.2f * v; }

template <int H>
__global__ void edge_max(const int* __restrict__ ei, const float* __restrict__ als,
                         const float* __restrict__ ald, float* __restrict__ m) {
  int e = blockIdx.x * blockDim.x + threadIdx.x;
  if (e >= ET) return;
  int s, d; edge_ends(ei, e, s, d);
#pragma unroll
  for (int h = 0; h < H; ++h)
    atomicMaxF(&m[d * H + h], leaky02(als[s * H + h] + ald[d * H + h]));
}

template <int H>
__global__ void edge_expsum(const int* __restrict__ ei, const float* __restrict__ als,
                            const float* __restrict__ ald, const float* __restrict__ m,
                            float* __restrict__ ssum) {
  int e = blockIdx.x * blockDim.x + threadIdx.x;
  if (e >= ET) return;
  int s, d; edge_ends(ei, e, s, d);
#pragma unroll
  for (int h = 0; h < H; ++h) {
    float v = leaky02(als[s * H + h] + ald[d * H + h]);
    atomicAdd(&ssum[d * H + h], expf(v - m[d * H + h]));
  }
}

template <int H, int C>
__global__ void edge_msg(const int* __restrict__ ei, const float* __restrict__ als,
                         const float* __restrict__ ald, const float* __restrict__ m,
                         const float* __restrict__ ssum, const float* __restrict__ hfeat,
                         float* __restrict__ agg, float* __restrict__ aval) {
  int e = blockIdx.x * blockDim.x + threadIdx.x;
  if (e >= ET) return;
  int s, d; edge_ends(ei, e, s, d);
  float asum = 0.f;
#pragma unroll
  for (int h = 0; h < H; ++h) {
    float v = leaky02(als[s * H + h] + ald[d * H + h]);
    float alpha = expf(v - m[d * H + h]) / ssum[d * H + h];
    asum += alpha;
#pragma unroll
    for (int c = 0; c < C; ++c)
      atomicAdd(&agg[(size_t)d * H * C + h * C + c],
                alpha * hfeat[(size_t)s * H * C + h * C + c]);
  }
  if (e < N_EDGES) aval[e] = asum * (1.0f / H);
}

// out = layernorm(elu(agg + bias)) * g + be  [+ res]
template <int D, bool HAS_RES>
__global__ void finalize_layer(const float* __restrict__ agg, const float* __restrict__ bias,
                               const float* __restrict__ g, const float* __restrict__ be,
                               const float* __restrict__ res, float* __restrict__ out) {
  int n = blockIdx.x * blockDim.x + threadIdx.x;
  if (n >= N_NODES) return;
  float v[D];
  float mu = 0.f;
#pragma unroll
  for (int i = 0; i < D; ++i) {
    float t = agg[(size_t)n * D + i] + bias[i];
    t = t > 0.f ? t : expm1f(t);
    v[i] = t;
    mu += t;
  }
  mu *= (1.0f / D);
  float var = 0.f;
#pragma unroll
  for (int i = 0; i < D; ++i) { float c = v[i] - mu; var += c * c; }
  var *= (1.0f / D);
  float r = rsqrtf(var + 1e-5f);
#pragma unroll
  for (int i = 0; i < D; ++i) {
    float o = (v[i] - mu) * r * g[i] + be[i];
    if (HAS_RES) o += res[(size_t)n * D + i];
    out[(size_t)n * D + i] = o;
  }
}

// ---------------------------------------------------------------------------
// Softmax over E edge values (values in (0,1], so no max-shift needed)
// ---------------------------------------------------------------------------
__global__ void reduce_expsum(const float* __restrict__ v, int n, float* __restrict__ accum) {
  __shared__ float sm[256];
  int i = blockIdx.x * 256 + threadIdx.x;
  sm[threadIdx.x] = (i < n) ? expf(v[i]) : 0.f;
  __syncthreads();
  for (int s = 128; s > 0; s >>= 1) {
    if (threadIdx.x < s) sm[threadIdx.x] += sm[threadIdx.x + s];
    __syncthreads();
  }
  if (threadIdx.x == 0) atomicAdd(accum, sm[0]);
}

__global__ void softmax_norm(float* __restrict__ v, int n, const float* __restrict__ accum) {
  int i = blockIdx.x * blockDim.x + threadIdx.x;
  if (i < n) v[i] = expf(v[i]) / (*accum);
}

// ---------------------------------------------------------------------------
extern "C" void kernel_launch(void* const* d_in, const int* in_sizes, int n_in,
                              void* d_out, int out_size, void* d_ws, size_t ws_size,
                              hipStream_t stream) {
  const float* x   = (const float*)d_in[0];
  const int*   ei  = (const int*)d_in[1];
  const float* W1  = (const float*)d_in[3];
  const float* as1 = (const float*)d_in[4];
  const float* ad1 = (const float*)d_in[5];
  const float* b1  = (const float*)d_in[6];
  const float* W2  = (const float*)d_in[7];
  const float* as2 = (const float*)d_in[8];
  const float* ad2 = (const float*)d_in[9];
  const float* b2  = (const float*)d_in[10];
  const float* W3  = (const float*)d_in[11];
  const float* as3 = (const float*)d_in[12];
  const float* ad3 = (const float*)d_in[13];
  const float* b3  = (const float*)d_in[14];
  const float* A1w = (const float*)d_in[15];
  const float* A1b = (const float*)d_in[16];
  const float* A2w = (const float*)d_in[17];
  const float* A2b = (const float*)d_in[18];
  const float* g1  = (const float*)d_in[19];
  const float* be1 = (const float*)d_in[20];
  const float* g2  = (const float*)d_in[21];
  const float* be2 = (const float*)d_in[22];
  const float* g3  = (const float*)d_in[23];
  const float* be3 = (const float*)d_in[24];

  float* ws   = (float*)d_ws;
  float* bufA = ws;                                  // N*64: h1pre -> h1 -> h2
  float* bufB = bufA + (size_t)N_NODES * 64;         // N*64: res1 | h2pre+res2 | h3pre
  float* bufC = bufB + (size_t)N_NODES * 64;         // N*64: agg
  float* als  = bufC + (size_t)N_NODES * 64;         // N*16
  float* ald  = als  + (size_t)N_NODES * 16;
  float* mbuf = ald  + (size_t)N_NODES * 16;
  float* sbuf = mbuf + (size_t)N_NODES * 16;
  float* sums = sbuf + (size_t)N_NODES * 16;         // 3 scalars

  float* out_h3 = (float*)d_out;
  float* out_e1 = out_h3 + (size_t)N_NODES * 16;
  float* out_e2 = out_e1 + N_EDGES;
  float* out_e3 = out_e2 + N_EDGES;

  const int MT = N_NODES / 16;                 // 4096 row tiles
  const int EG = (ET + 255) / 256;             // edge-pass grid
  const int NG = (N_NODES + 255) / 256;        // node-pass grid
  auto fg = [](int n) { return (n + 255) / 256; };
  const float NEG_INF = -__builtin_huge_valf();

  fill_kernel<<<1, 256, 0, stream>>>(sums, 3, 0.f);

  // ---------------- Layer 1: 84 -> 16 heads x 4 ----------------
  gemm16_wmma<<<MT * 4, 32, 0, stream>>>(x, W1, nullptr, bufA, 84, 64);
  gemm16_wmma<<<MT * 4, 32, 0, stream>>>(x, A1w, A1b, bufB, 84, 64);
  node_coeff<16, 4><<<fg(N_NODES * 16), 256, 0, stream>>>(bufA, as1, ad1, als, ald);
  fill_kernel<<<fg(N_NODES * 16), 256, 0, stream>>>(mbuf, N_NODES * 16, NEG_INF);
  fill_kernel<<<fg(N_NODES * 16), 256, 0, stream>>>(sbuf, N_NODES * 16, 0.f);
  fill_kernel<<<fg(N_NODES * 64), 256, 0, stream>>>(bufC, N_NODES * 64, 0.f);
  edge_max<16><<<EG, 256, 0, stream>>>(ei, als, ald, mbuf);
  edge_expsum<16><<<EG, 256, 0, stream>>>(ei, als, ald, mbuf, sbuf);
  edge_msg<16, 4><<<EG, 256, 0, stream>>>(ei, als, ald, mbuf, sbuf, bufA, bufC, out_e1);
  finalize_layer<64, true><<<NG, 256, 0, stream>>>(bufC, b1, g1, be1, bufB, bufA);  // h1

  // ---------------- Layer 2: 64 -> 8 heads x 4 ----------------
  float* h2pre = bufB;
  float* res2  = bufB + (size_t)N_NODES * 32;
  gemm16_wmma<<<MT * 2, 32, 0, stream>>>(bufA, W2, nullptr, h2pre, 64, 32);
  gemm16_wmma<<<MT * 2, 32, 0, stream>>>(bufA, A2w, A2b, res2, 64, 32);
  node_coeff<8, 4><<<fg(N_NODES * 8), 256, 0, stream>>>(h2pre, as2, ad2, als, ald);
  fill_kernel<<<fg(N_NODES * 8), 256, 0, stream>>>(mbuf, N_NODES * 8, NEG_INF);
  fill_kernel<<<fg(N_NODES * 8), 256, 0, stream>>>(sbuf, N_NODES * 8, 0.f);
  fill_kernel<<<fg(N_NODES * 32), 256, 0, stream>>>(bufC, N_NODES * 32, 0.f);
  edge_max<8><<<EG, 256, 0, stream>>>(ei, als, ald, mbuf);
  edge_expsum<8><<<EG, 256, 0, stream>>>(ei, als, ald, mbuf, sbuf);
  edge_msg<8, 4><<<EG, 256, 0, stream>>>(ei, als, ald, mbuf, sbuf, h2pre, bufC, out_e2);
  finalize_layer<32, true><<<NG, 256, 0, stream>>>(bufC, b2, g2, be2, res2, bufA);  // h2

  // ---------------- Layer 3: 32 -> 4 heads x 4 ----------------
  float* h3pre = bufB;
  gemm16_wmma<<<MT * 1, 32, 0, stream>>>(bufA, W3, nullptr, h3pre, 32, 16);
  node_coeff<4, 4><<<fg(N_NODES * 4), 256, 0, stream>>>(h3pre, as3, ad3, als, ald);
  fill_kernel<<<fg(N_NODES * 4), 256, 0, stream>>>(mbuf, N_NODES * 4, NEG_INF);
  fill_kernel<<<fg(N_NODES * 4), 256, 0, stream>>>(sbuf, N_NODES * 4, 0.f);
  fill_kernel<<<fg(N_NODES * 16), 256, 0, stream>>>(bufC, N_NODES * 16, 0.f);
  edge_max<4><<<EG, 256, 0, stream>>>(ei, als, ald, mbuf);
  edge_expsum<4><<<EG, 256, 0, stream>>>(ei, als, ald, mbuf, sbuf);
  edge_msg<4, 4><<<EG, 256, 0, stream>>>(ei, als, ald, mbuf, sbuf, h3pre, bufC, out_e3);
  finalize_layer<16, false><<<NG, 256, 0, stream>>>(bufC, b3, g3, be3, nullptr, out_h3);

  // ---------------- Edge softmaxes e1..e3 ----------------
  reduce_expsum<<<fg(N_EDGES), 256, 0, stream>>>(out_e1, N_EDGES, &sums[0]);
  softmax_norm<<<fg(N_EDGES), 256, 0, stream>>>(out_e1, N_EDGES, &sums[0]);
  reduce_expsum<<<fg(N_EDGES), 256, 0, stream>>>(out_e2, N_EDGES, &sums[1]);
  softmax_norm<<<fg(N_EDGES), 256, 0, stream>>>(out_e2, N_EDGES, &sums[1]);
  reduce_expsum<<<fg(N_EDGES), 256, 0, stream>>>(out_e3, N_EDGES, &sums[2]);
  softmax_norm<<<fg(N_EDGES), 256, 0, stream>>>(out_e3, N_EDGES, &sums[2]);
}